// Sum_54717883351445
// MI455X (gfx1250) — compile-verified
//
#include <hip/hip_runtime.h>

// Problem constants (from reference): B=128, F=512, IC=32, OC=32
#define B_DIM 128
#define F_DIM 512
#define IC_DIM 32
#define OC_DIM 32
#define PAD 34   // even (8B-aligned float2 reads) and conflict-free stride mod 64 banks

typedef __attribute__((ext_vector_type(2))) float v2f;
typedef __attribute__((ext_vector_type(8))) float v8f;

struct __align__(16) Smem {
    float E[B_DIM * PAD];      // exp(x - rowmax), [b][ic] padded
    float SMT[OC_DIM * PAD];   // softmax(W) transposed: [oc][ic] padded
    float W[IC_DIM * OC_DIM];  // raw weight staging [ic][oc]
    float xmax[B_DIM];
};

__global__ __launch_bounds__(128) void fused_lse_wmma_kernel(
    const float* __restrict__ x,      // [B, F, IC]
    const float* __restrict__ w,      // [F, IC, OC]
    float* __restrict__ out)          // [B, F, OC]
{
    __shared__ Smem sm;
    const int f = blockIdx.x;
    const int t = threadIdx.x;

    // ---- Phase A1: stage W[f] (32x32 f32 = 1024 floats) into LDS, coalesced b128 ----
    {
        const float4* wsrc = (const float4*)(w + (size_t)f * IC_DIM * OC_DIM);
        float4* wdst = (float4*)sm.W;
        wdst[t]       = wsrc[t];        // 128 threads x 2 float4 = 256 float4
        wdst[t + 128] = wsrc[t + 128];
    }
    __syncthreads();

    // ---- Phase A2: wave 0 computes softmax over IC per oc-column, stores transposed ----
    if (t < 32) {
        const int oc = t;
        float m = -INFINITY;
        #pragma unroll
        for (int ic = 0; ic < IC_DIM; ++ic)
            m = fmaxf(m, sm.W[ic * OC_DIM + oc]);
        float s = 0.0f;
        #pragma unroll
        for (int ic = 0; ic < IC_DIM; ++ic)
            s += __expf(sm.W[ic * OC_DIM + oc] - m);
        const float inv = 1.0f / s;
        #pragma unroll
        for (int ic = 0; ic < IC_DIM; ++ic)
            sm.SMT[oc * PAD + ic] = __expf(sm.W[ic * OC_DIM + oc] - m) * inv;
    }

    // ---- Phase B: one x-row per thread -> rowmax + exp(x - rowmax) into LDS ----
    {
        const int b = t;  // 128 threads, 128 rows
        const float4* xr = (const float4*)(x + ((size_t)b * F_DIM + f) * IC_DIM);
        float4 v[8];
        #pragma unroll
        for (int i = 0; i < 8; ++i) v[i] = xr[i];
        float m = -INFINITY;
        #pragma unroll
        for (int i = 0; i < 8; ++i)
            m = fmaxf(m, fmaxf(fmaxf(v[i].x, v[i].y), fmaxf(v[i].z, v[i].w)));
        sm.xmax[b] = m;
        float* er = &sm.E[b * PAD];
        #pragma unroll
        for (int i = 0; i < 8; ++i) {
            er[4 * i + 0] = __expf(v[i].x - m);
            er[4 * i + 1] = __expf(v[i].y - m);
            er[4 * i + 2] = __expf(v[i].z - m);
            er[4 * i + 3] = __expf(v[i].w - m);
        }
    }
    __syncthreads();

    // ---- Phase C: per-wave 32x32 GEMM block via V_WMMA_F32_16X16X4_F32 ----
    // A (16x4 f32): lane<16 -> rows 0..15, K={0,1}; lane>=16 -> K={2,3}. float2 per lane.
    // B (4x16 f32): lane<16 -> N=lane, K={0,1}; lane>=16 -> N=lane-16, K={2,3}.
    const int wave  = t >> 5;
    const int lane  = t & 31;
    const int lhalf = lane >> 4;      // 0 or 1
    const int lmod  = lane & 15;
    const int kb    = lhalf * 2;      // K sub-offset for this lane
    const int mbase = wave * 32;      // batch rows owned by this wave

    v8f acc[2][2] = {};               // [mtile][ntile], each 16x16 f32

    #pragma unroll
    for (int ks = 0; ks < 8; ++ks) {  // K blocks of 4 over IC=32
        const int k0 = ks * 4 + kb;
        v2f bfrag[2];
        #pragma unroll
        for (int nt = 0; nt < 2; ++nt) {
            const int n = nt * 16 + lmod;
            bfrag[nt] = *(const v2f*)&sm.SMT[n * PAD + k0];
        }
        #pragma unroll
        for (int mt = 0; mt < 2; ++mt) {
            const int row = mbase + mt * 16 + lmod;
            const v2f afrag = *(const v2f*)&sm.E[row * PAD + k0];
            #pragma unroll
            for (int nt = 0; nt < 2; ++nt) {
                // 8 args: (neg_a, A, neg_b, B, c_mod, C, reuse_a, reuse_b)
                acc[mt][nt] = __builtin_amdgcn_wmma_f32_16x16x4_f32(
                    false, afrag, false, bfrag[nt], (short)0, acc[mt][nt],
                    false, false);
            }
        }
    }

    // ---- Phase D: out = log(dot) + xmax, per C/D f32 16x16 layout ----
    // VGPR r holds row (r + 8*lhalf) of the tile; N = lmod within tile.
    #pragma unroll
    for (int mt = 0; mt < 2; ++mt) {
        #pragma unroll
        for (int nt = 0; nt < 2; ++nt) {
            #pragma unroll
            for (int r = 0; r < 8; ++r) {
                const int b  = mbase + mt * 16 + r + 8 * lhalf;
                const int oc = nt * 16 + lmod;
                out[((size_t)b * F_DIM + f) * OC_DIM + oc] =
                    __logf(acc[mt][nt][r]) + sm.xmax[b];
            }
        }
    }
}

extern "C" void kernel_launch(void* const* d_in, const int* in_sizes, int n_in,
                              void* d_out, int out_size, void* d_ws, size_t ws_size,
                              hipStream_t stream) {
    (void)in_sizes; (void)n_in; (void)out_size; (void)d_ws; (void)ws_size;
    const float* x = (const float*)d_in[0];        // [128, 512, 32] f32
    const float* w = (const float*)d_in[1];        // [512, 32, 32] f32
    float* out     = (float*)d_out;                // [128, 512, 32] f32
    fused_lse_wmma_kernel<<<dim3(F_DIM), dim3(128), 0, stream>>>(x, w, out);
}